// CameraExtrinsics_70007966924875
// MI455X (gfx1250) — compile-verified
//
#include <hip/hip_runtime.h>
#include <math.h>

// Table entry: R row-major [9] + t [3] = 12 floats = 48 bytes (16B-aligned stride).
#define TBL_STRIDE 12
#define MAX_IMAGES 1024

// ---- CDNA5 async global->LDS copy (gfx1250, ASYNCcnt path), inline asm ----
// ISA §15.18 VGLOBAL: GLOBAL_LOAD_ASYNC_TO_LDS_B128, GVS mode:
//   global address is the 64-bit SGPR base plus the 32-bit VGPR offset; LDS address comes from vdst
// Asm operand order (dst first, matching all AMDGPU loads): vdst(lds), vaddr, saddr.
__device__ __forceinline__ void async_copy_b128(const void* gbase, unsigned byte_off,
                                                void* lds_dst) {
    unsigned lds_off = (unsigned)(size_t)lds_dst;  // low 32 bits of flat = LDS offset
    asm volatile("global_load_async_to_lds_b128 %0, %1, %2"
                 :: "v"(lds_off), "v"(byte_off), "s"(gbase)
                 : "memory");
}

__device__ __forceinline__ void wait_asynccnt0() {
    asm volatile("s_wait_asynccnt 0" ::: "memory");
}

// ---------------------------------------------------------------------------
// Kernel 1: compute Rodrigues rotation + pack [R(9) | t(3)] per image.
// Runs once over n_images (=1000) -> transcendentals amortized off the
// 4.2M-ray hot path entirely.
// ---------------------------------------------------------------------------
__global__ void rodrigues_table_kernel(const float* __restrict__ rot,
                                       const float* __restrict__ trans,
                                       float* __restrict__ table,
                                       int n_images) {
    int idx = blockIdx.x * blockDim.x + threadIdx.x;
    if (idx >= n_images) return;
    float rx = rot[3 * idx + 0];
    float ry = rot[3 * idx + 1];
    float rz = rot[3 * idx + 2];
    float th2 = rx * rx + ry * ry + rz * rz;
    float th  = sqrtf(th2);
    float a, b;  // sin_c, cos_c
    if (th < 1e-6f) {
        a = 1.0f - th2 * (1.0f / 6.0f);
        b = 0.5f - th2 * (1.0f / 24.0f);
    } else {
        a = sinf(th) / th;
        b = (1.0f - cosf(th)) / th2;
    }
    float* e = table + (size_t)idx * TBL_STRIDE;
    // R = I + a*K + b*K^2, K = skew(r)
    e[0]  = 1.0f - b * (ry * ry + rz * rz);
    e[1]  = -a * rz + b * rx * ry;
    e[2]  =  a * ry + b * rx * rz;
    e[3]  =  a * rz + b * rx * ry;
    e[4]  = 1.0f - b * (rx * rx + rz * rz);
    e[5]  = -a * rx + b * ry * rz;
    e[6]  = -a * ry + b * rx * rz;
    e[7]  =  a * rx + b * ry * rz;
    e[8]  = 1.0f - b * (rx * rx + ry * ry);
    e[9]  = trans[3 * idx + 0];
    e[10] = trans[3 * idx + 1];
    e[11] = trans[3 * idx + 2];
}

// ---------------------------------------------------------------------------
// Per-ray math from a gathered table entry (3x float4, 16B aligned).
// ---------------------------------------------------------------------------
__device__ __forceinline__ void do_ray(const float4 R0, const float4 R1, const float4 R2,
                                       const float* __restrict__ o,
                                       const float* __restrict__ d,
                                       float* __restrict__ out_o,
                                       float* __restrict__ out_d,
                                       float* __restrict__ out_R,
                                       float* __restrict__ out_t,
                                       long long r) {
    // R0 = R00 R01 R02 R10 ; R1 = R11 R12 R20 R21 ; R2 = R22 t0 t1 t2
    float ox = o[3 * r + 0], oy = o[3 * r + 1], oz = o[3 * r + 2];
    float dx = d[3 * r + 0], dy = d[3 * r + 1], dz = d[3 * r + 2];

    float no0 = fmaf(R0.x, ox, fmaf(R0.y, oy, fmaf(R0.z, oz, R2.y)));
    float no1 = fmaf(R0.w, ox, fmaf(R1.x, oy, fmaf(R1.y, oz, R2.z)));
    float no2 = fmaf(R1.z, ox, fmaf(R1.w, oy, fmaf(R2.x, oz, R2.w)));

    float nd0 = fmaf(R0.x, dx, fmaf(R0.y, dy, R0.z * dz));
    float nd1 = fmaf(R0.w, dx, fmaf(R1.x, dy, R1.y * dz));
    float nd2 = fmaf(R1.z, dx, fmaf(R1.w, dy, R2.x * dz));

    out_o[3 * r + 0] = no0; out_o[3 * r + 1] = no1; out_o[3 * r + 2] = no2;
    out_d[3 * r + 0] = nd0; out_d[3 * r + 1] = nd1; out_d[3 * r + 2] = nd2;

    float* Rp = out_R + 9 * r;
    Rp[0] = R0.x; Rp[1] = R0.y; Rp[2] = R0.z;
    Rp[3] = R0.w; Rp[4] = R1.x; Rp[5] = R1.y;
    Rp[6] = R1.z; Rp[7] = R1.w; Rp[8] = R2.x;

    out_t[3 * r + 0] = R2.y; out_t[3 * r + 1] = R2.z; out_t[3 * r + 2] = R2.w;
}

// ---------------------------------------------------------------------------
// Kernel 2: stage table -> LDS via gfx1250 async copies (ASYNCcnt path),
// then grid-stride stream over rays with LDS gathers (ds_load_b128 x3).
// ---------------------------------------------------------------------------
__global__ void __launch_bounds__(256)
transform_kernel(const int* __restrict__ img_idx,
                 const float* __restrict__ o,
                 const float* __restrict__ d,
                 const float* __restrict__ table,
                 float* __restrict__ out,
                 int n_rays, int n_images) {
    __shared__ __align__(16) float lut[MAX_IMAGES * TBL_STRIDE];

    const long long N = n_rays;
    float* out_o = out;
    float* out_d = out + 3LL * N;
    float* out_R = out + 6LL * N;
    float* out_t = out + 15LL * N;

    const int gtid = blockIdx.x * blockDim.x + threadIdx.x;
    const int nth  = gridDim.x * blockDim.x;

    const bool use_lds = (n_images <= MAX_IMAGES);

    if (use_lds) {
        // Async global->LDS staging of the 48KB table, one 16B chunk per lane.
        const int nchunks = (n_images * TBL_STRIDE) / 4;  // float4 chunks
        for (int c = threadIdx.x; c < nchunks; c += blockDim.x) {
            async_copy_b128(table, 16u * (unsigned)c, (char*)lut + 16u * (unsigned)c);
        }
        wait_asynccnt0();
    }
    __syncthreads();

    if (use_lds) {
        for (long long r = gtid; r < N; r += nth) {
            int img = img_idx[r];
            const float* e = &lut[img * TBL_STRIDE];
            float4 R0 = *(const float4*)(e + 0);
            float4 R1 = *(const float4*)(e + 4);
            float4 R2 = *(const float4*)(e + 8);
            do_ray(R0, R1, R2, o, d, out_o, out_d, out_R, out_t, r);
        }
    } else {
        // Fallback: gather straight from the L2-resident global table.
        for (long long r = gtid; r < N; r += nth) {
            int img = img_idx[r];
            const float* e = table + (size_t)img * TBL_STRIDE;
            float4 R0 = *(const float4*)(e + 0);
            float4 R1 = *(const float4*)(e + 4);
            float4 R2 = *(const float4*)(e + 8);
            do_ray(R0, R1, R2, o, d, out_o, out_d, out_R, out_t, r);
        }
    }
}

// ---------------------------------------------------------------------------
// Launch: inputs (setup_inputs order): i, o, d, rotation, translation.
// Output: [new_o 3N | new_d 3N | R 9N | t 3N] floats.
// ---------------------------------------------------------------------------
extern "C" void kernel_launch(void* const* d_in, const int* in_sizes, int n_in,
                              void* d_out, int out_size, void* d_ws, size_t ws_size,
                              hipStream_t stream) {
    const int*   i_ptr = (const int*)d_in[0];
    const float* o_ptr = (const float*)d_in[1];
    const float* d_ptr = (const float*)d_in[2];
    const float* rot   = (const float*)d_in[3];
    const float* trans = (const float*)d_in[4];

    int n_rays   = in_sizes[0];
    int n_images = in_sizes[3] / 3;

    float* table = (float*)d_ws;  // n_images * 12 floats (48KB for 1000 images)

    int tb = (n_images + 255) / 256;
    rodrigues_table_kernel<<<tb, 256, 0, stream>>>(rot, trans, table, n_images);

    // 1024 blocks: 8192 waves (plenty to saturate 23.3 TB/s HBM) while keeping
    // the per-block LDS staging replay down to ~48MB of L2 traffic total.
    int need = (n_rays + 255) / 256;
    int nblocks = need < 1024 ? need : 1024;
    transform_kernel<<<nblocks, 256, 0, stream>>>(i_ptr, o_ptr, d_ptr, table,
                                                  (float*)d_out, n_rays, n_images);
}